// ChebNet_36094905155904
// MI455X (gfx1250) — compile-verified
//
#include <hip/hip_runtime.h>

// ---------------------------------------------------------------------------
// ChebNet (K=3, sym-norm, lambda_max=2) for MI455X / gfx1250, wave32.
// Dense GEMMs: V_WMMA_F32_16X16X4_F32, M-blocked (5 row-tiles/wave), A tile
// staged in LDS (conflict-free padding), B via immediate-offset global loads.
// Sparse propagation: L2-resident f32 hardware scatter atomics.
// ---------------------------------------------------------------------------

typedef __attribute__((ext_vector_type(2))) float v2f;
typedef __attribute__((ext_vector_type(8))) float v8f;

#define HIDC 128        // IN_C == HID_C (leading dim of all Tx buffers)
#define MT   5          // M tiles (of 16 rows) per block: 80 rows/block
#define ROWS (MT * 16)  // 80 ; N=100000 = 1250 * 80 exactly
#define LDA  388        // padded LDS row stride (floats): 388%4==0 (16B align),
                        // 388%64==4 -> 32 lanes x b64 cover all 64 banks once

__device__ __forceinline__ void atomic_add_f32(float* p, float v) {
  __hip_atomic_fetch_add(p, v, __ATOMIC_RELAXED, __HIP_MEMORY_SCOPE_AGENT);
}

// ---------------- utility kernels ----------------
__global__ void zero_f32x4(float* __restrict__ p, long long n4) {
  long long i = (long long)blockIdx.x * blockDim.x + threadIdx.x;
  if (i < n4) ((float4*)p)[i] = make_float4(0.f, 0.f, 0.f, 0.f);
}

__global__ void deg_count(const long long* __restrict__ src,
                          float* __restrict__ deg, long long E) {
  long long e = (long long)blockIdx.x * blockDim.x + threadIdx.x;
  if (e < E) atomic_add_f32(&deg[(long long)src[e]], 1.0f);
}

__global__ void deg_to_dinv(float* __restrict__ d, int n) {
  int i = blockIdx.x * blockDim.x + threadIdx.x;
  if (i < n) {
    float v = d[i];
    d[i] = (v > 0.f) ? rsqrtf(v) : 0.f;
  }
}

// y[dst] += w_e * h[src],  w_e = -dinv[src]*dinv[dst].
// 32 lanes per edge, float4 of channels per lane (C == 128); channel accesses
// coalesced across the wave; h (51.2MB) is resident in the 192MB L2.
__global__ void prop_scatter(const long long* __restrict__ src,
                             const long long* __restrict__ dst,
                             const float* __restrict__ dinv,
                             const float* __restrict__ hin,
                             float* __restrict__ tout, long long E) {
  long long gid = (long long)blockIdx.x * blockDim.x + threadIdx.x;
  long long e = gid >> 5;
  int c4 = ((int)(gid & 31)) << 2;
  if (e >= E) return;
  long long s = src[e];
  long long d = dst[e];
  float w = -dinv[s] * dinv[d];
  const float4 h = *(const float4*)(hin + s * HIDC + c4);
  float* o = tout + d * HIDC + c4;
  atomic_add_f32(o + 0, w * h.x);
  atomic_add_f32(o + 1, w * h.y);
  atomic_add_f32(o + 2, w * h.z);
  atomic_add_f32(o + 3, w * h.w);
}

// t2 = 2*t2 - tx0   (Chebyshev recurrence Tx2 = 2*L_hat*Tx1 - Tx0)
__global__ void cheb_combine(float* __restrict__ t2,
                             const float* __restrict__ tx0, long long n4) {
  long long i = (long long)blockIdx.x * blockDim.x + threadIdx.x;
  if (i < n4) {
    float4 a = ((const float4*)t2)[i];
    float4 b = ((const float4*)tx0)[i];
    ((float4*)t2)[i] =
        make_float4(2.f * a.x - b.x, 2.f * a.y - b.y,
                    2.f * a.z - b.z, 2.f * a.w - b.w);
  }
}

// out = Tx0@W[0] + Tx1@W[1] + Tx2@W[2] + b (+ReLU), fp32 WMMA.
// Block: COUT/16 waves, each owns one 16-col slice for 5 row-tiles (80 rows).
// A tile (80 x 384) staged once per block into LDS (row stride LDA=388 floats,
// bank-conflict-free, 16B aligned). B loads: per-lane base + constant
// immediate offsets (COUT is a template constant). K = 3*128, stepped by 4.
//
// f32 16x16x4 operand layout (ISA 7.12.2):
//   A (16x4): lane<16 -> M=lane, v0=K0,v1=K1 ; lane>=16 -> v0=K2,v1=K3
//   B (4x16): lane<16 -> N=lane, v0=K0,v1=K1 ; lane>=16 -> v0=K2,v1=K3
//   C/D     : VGPR r, lane L -> row = r + 8*(L>=16), col = L%16
template <int COUT, int RELU>
__global__ __launch_bounds__(32 * (COUT / 16)) void cheb_gemm_wmma(
    const float* __restrict__ tx0, const float* __restrict__ tx1,
    const float* __restrict__ tx2, const float* __restrict__ W,  // [3,128,COUT]
    const float* __restrict__ bias,                              // [COUT]
    float* __restrict__ out,                                     // [N,COUT]
    int N) {
  extern __shared__ float sA[];  // [ROWS][LDA]

  const int tid  = threadIdx.x;
  const int lane = tid & 31;
  const int wave = tid >> 5;
  const int half = lane >> 4;
  const int l    = lane & 15;

  const long long m0 = (long long)blockIdx.x * ROWS;

  // ---- cooperative stage: A tile (concat Tx0|Tx1|Tx2 per row) -> LDS ----
  const float* Abufs[3] = {tx0, tx1, tx2};
  for (int idx = tid; idx < ROWS * 96; idx += blockDim.x) {  // 96 float4 / row
    int row   = idx / 96;
    int c4    = idx - row * 96;
    int chunk = c4 >> 5;   // which Tx buffer
    int k4    = c4 & 31;   // float4 within the 128-wide chunk
    long long gm = m0 + row;
    if (gm > (long long)N - 1) gm = (long long)N - 1;
    float4 v = *(const float4*)(Abufs[chunk] + gm * HIDC + k4 * 4);
    *(float4*)(sA + (size_t)row * LDA + c4 * 4) = v;
  }
  __syncthreads();

  const int n = wave * 16 + l;  // output column this lane owns

  v8f acc[MT];
#pragma unroll
  for (int mt = 0; mt < MT; ++mt) acc[mt] = (v8f){0.f, 0.f, 0.f, 0.f,
                                                  0.f, 0.f, 0.f, 0.f};

#pragma unroll
  for (int chunk = 0; chunk < 3; ++chunk) {
    // per-lane bases; everything below is constant immediate offsets
    const float* Bb = W + ((size_t)chunk * HIDC + half * 2) * COUT + n;
    const float* As = sA + chunk * HIDC + half * 2 + (size_t)l * LDA;
#pragma unroll 8
    for (int k0 = 0; k0 < HIDC; k0 += 4) {
      v2f b;
      b.x = Bb[(size_t)k0 * COUT];
      b.y = Bb[(size_t)k0 * COUT + COUT];
#pragma unroll
      for (int mt = 0; mt < MT; ++mt) {
        v2f a = *(const v2f*)(As + (size_t)mt * 16 * LDA + k0);
        acc[mt] = __builtin_amdgcn_wmma_f32_16x16x4_f32(
            /*neg_a=*/false, a, /*neg_b=*/false, b,
            /*c_mod=*/(short)0, acc[mt], /*reuse_a=*/false, /*reuse_b=*/false);
      }
    }
  }

  // ---- write back ----
  const float bv = bias[n];
#pragma unroll
  for (int mt = 0; mt < MT; ++mt) {
    const long long rbase = m0 + mt * 16 + half * 8;
    float* op = out + rbase * COUT + n;
#pragma unroll
    for (int r = 0; r < 8; ++r) {
      if (rbase + r < N) {
        float v = acc[mt][r] + bv;
        if (RELU) v = fmaxf(v, 0.f);
        op[(size_t)r * COUT] = v;
      }
    }
  }
}

// ---------------------------------------------------------------------------
static inline size_t align256(size_t x) { return (x + 255) & ~(size_t)255; }

template <int COUT, int RELU>
static void launch_gemm(const float* h, const float* t1, const float* t2,
                        const float* Wl, const float* bl, float* o, int N,
                        hipStream_t stream) {
  dim3 grid((N + ROWS - 1) / ROWS);
  dim3 block(32 * (COUT / 16));
  size_t lds = (size_t)ROWS * LDA * sizeof(float);
  cheb_gemm_wmma<COUT, RELU><<<grid, block, lds, stream>>>(h, t1, t2, Wl, bl,
                                                           o, N);
}

static void run_props(const float* hin, float* t1, float* t2, float* dinv,
                      const long long* src, const long long* dst, int N,
                      long long E, hipStream_t stream) {
  const long long feat4 = (long long)N * HIDC / 4;
  const int TB = 256;
  const unsigned zb = (unsigned)((feat4 + TB - 1) / TB);
  const unsigned pb = (unsigned)((E * 32 + TB - 1) / TB);
  zero_f32x4<<<zb, TB, 0, stream>>>(t1, feat4);
  prop_scatter<<<pb, TB, 0, stream>>>(src, dst, dinv, hin, t1, E);
  zero_f32x4<<<zb, TB, 0, stream>>>(t2, feat4);
  prop_scatter<<<pb, TB, 0, stream>>>(src, dst, dinv, t1, t2, E);
  cheb_combine<<<zb, TB, 0, stream>>>(t2, hin, feat4);
}

extern "C" void kernel_launch(void* const* d_in, const int* in_sizes, int n_in,
                              void* d_out, int out_size, void* d_ws,
                              size_t ws_size, hipStream_t stream) {
  const float*     x  = (const float*)d_in[0];
  const long long* ei = (const long long*)d_in[1];  // int64 [2, E]
  const float* W0 = (const float*)d_in[2];
  const float* b0 = (const float*)d_in[3];
  const float* W1 = (const float*)d_in[4];
  const float* b1 = (const float*)d_in[5];
  const float* W2 = (const float*)d_in[6];
  const float* b2 = (const float*)d_in[7];
  float* out = (float*)d_out;

  const int       N = in_sizes[0] / HIDC;          // 100000
  const long long E = (long long)in_sizes[1] / 2;  // 1.6M

  const long long* src = ei;
  const long long* dst = ei + E;

  // workspace carve-up
  char* ws = (char*)d_ws;
  size_t off = 0;
  float* dinv = (float*)(ws + off); off += align256((size_t)N * 4);
  const size_t featB = align256((size_t)N * HIDC * sizeof(float));
  float* hA = (float*)(ws + off); off += featB;
  float* hB = (float*)(ws + off); off += featB;
  float* t1 = (float*)(ws + off); off += featB;
  float* t2 = (float*)(ws + off); off += featB;

  const int TB = 256;

  // deg (counted over src, per reference) -> dinv
  zero_f32x4<<<(unsigned)((N / 4 + TB - 1) / TB), TB, 0, stream>>>(
      dinv, (long long)N / 4);
  deg_count<<<(unsigned)((E + TB - 1) / TB), TB, 0, stream>>>(src, dinv, E);
  deg_to_dinv<<<(unsigned)((N + TB - 1) / TB), TB, 0, stream>>>(dinv, N);

  // layer 0: x -> hA (relu)
  run_props(x, t1, t2, dinv, src, dst, N, E, stream);
  launch_gemm<HIDC, 1>(x, t1, t2, W0, b0, hA, N, stream);
  // layer 1: hA -> hB (relu)
  run_props(hA, t1, t2, dinv, src, dst, N, E, stream);
  launch_gemm<HIDC, 1>(hA, t1, t2, W1, b1, hB, N, stream);
  // layer 2: hB -> out
  run_props(hB, t1, t2, dinv, src, dst, N, E, stream);
  launch_gemm<64, 0>(hB, t1, t2, W2, b2, out, N, stream);
}